// WDTransformer_51780125721356
// MI455X (gfx1250) — compile-verified
//
#include <hip/hip_runtime.h>
#include <stdint.h>

// Problem constants (hardcoded in the reference).
constexpr int kB  = 4;
constexpr int kC  = 3;
constexpr int kH  = 436;
constexpr int kW  = 1024;
constexpr int kHp = 435;   // kH - 1
constexpr int kWp = 1023;  // kW - 1

// Tiling: 16x64 output pixels per 256-thread block (8 wave32s).
constexpr int kTI  = 16;
constexpr int kTJ  = 64;
constexpr int kHR  = kTI + 2;  // 18 halo rows
constexpr int kHC  = kTJ + 2;  // 66 halo cols
constexpr int kLDW = 68;       // padded LDS row stride
constexpr int kNTH = 256;

__device__ static inline unsigned lds_offset(const void* p) {
    // Low 32 bits of a generic pointer to LDS are the wave-relative LDS offset.
    return (unsigned)(uintptr_t)p;
}

__global__ __launch_bounds__(kNTH) void wdt_flow_kernel(
    const float* __restrict__ x, const float* __restrict__ y,
    float* __restrict__ out) {
    __shared__ float sx[kC][kHR][kLDW];
    __shared__ float sy[kC][kHR][kLDW];

    const int b   = blockIdx.z;
    const int I0  = blockIdx.y * kTI;
    const int J0  = blockIdx.x * kTJ;
    const int tid = threadIdx.x;

    const float* xb = x + (size_t)b * kC * kH * kW;
    const float* yb = y + (size_t)b * kC * kH * kW;

    // ---- Stage halo tiles into LDS via CDNA5 async global->LDS loads ----
    for (int idx = tid; idx < kC * kHR * kHC; idx += kNTH) {
        int ch  = idx / (kHR * kHC);
        int rem = idx - ch * (kHR * kHC);
        int r   = rem / kHC;
        int c   = rem - r * kHC;
        int gr  = I0 - 1 + r;
        int gc  = J0 - 1 + c;
        unsigned lx = lds_offset(&sx[ch][r][c]);
        unsigned ly = lds_offset(&sy[ch][r][c]);
        if (gr >= 0 && gr < kH && gc >= 0 && gc < kW) {
            unsigned long long gax =
                (unsigned long long)(uintptr_t)(xb + ((size_t)ch * kH + gr) * kW + gc);
            unsigned long long gay =
                (unsigned long long)(uintptr_t)(yb + ((size_t)ch * kH + gr) * kW + gc);
            asm volatile("global_load_async_to_lds_b32 %0, %1, off"
                         :: "v"(lx), "v"(gax) : "memory");
            asm volatile("global_load_async_to_lds_b32 %0, %1, off"
                         :: "v"(ly), "v"(gay) : "memory");
        } else {
            sx[ch][r][c] = 0.0f;  // zero halo beyond image; never used by valid terms
            sy[ch][r][c] = 0.0f;
        }
    }
#if __has_builtin(__builtin_amdgcn_s_wait_asynccnt)
    __builtin_amdgcn_s_wait_asynccnt(0);
#else
    asm volatile("s_wait_asynccnt 0" ::: "memory");
#endif
    __syncthreads();

    // ---- Compute: gather-style fold, 4 pixels per thread ----
    const int tx = tid & (kTJ - 1);
    const int ty = tid >> 6;  // 0..3
    float* outu = out + ((size_t)b * 2 + 0) * kH * kW;
    float* outv = out + ((size_t)b * 2 + 1) * kH * kW;

    #pragma unroll
    for (int k = 0; k < 4; ++k) {
        const int I = I0 + ty * 4 + k;
        const int J = J0 + tx;            // kW % kTJ == 0, always in range
        if (I >= kH) continue;
        const int li = I - I0 + 1;        // 1..16
        const int lj = tx + 1;            // 1..64

        const float xc0 = sx[0][li][lj];
        const float xc1 = sx[1][li][lj];
        const float xc2 = sx[2][li][lj];
        const float nx  = sqrtf(xc0 * xc0 + xc1 * xc1 + xc2 * xc2);

        // cosine(x(I,J), -y(I-1+r, J-1+c)) table over the 3x3 y-neighborhood
        float cosn[3][3];
        #pragma unroll
        for (int r = 0; r < 3; ++r) {
            #pragma unroll
            for (int c = 0; c < 3; ++c) {
                const float y0 = sy[0][li - 1 + r][lj - 1 + c];
                const float y1 = sy[1][li - 1 + r][lj - 1 + c];
                const float y2 = sy[2][li - 1 + r][lj - 1 + c];
                const float dot = xc0 * y0 + xc1 * y1 + xc2 * y2;
                const float ny  = sqrtf(y0 * y0 + y1 * y1 + y2 * y2);
                cosn[r][c] = -dot / fmaxf(nx * ny, 1e-8f);
            }
        }

        float sdx = 0.0f, sdy = 0.0f;
        #pragma unroll
        for (int pi = 0; pi < 2; ++pi) {
            #pragma unroll
            for (int pj = 0; pj < 2; ++pj) {
                const int h = I - pi, w = J - pj;
                if (h < 0 || h >= kHp || w < 0 || w >= kWp) continue;
                // q=(qr,qc) -> cosn[1-pi+qr][1-pj+qc]
                const float c00 = cosn[1 - pi][1 - pj];
                const float c01 = cosn[1 - pi][2 - pj];
                const float c10 = cosn[2 - pi][1 - pj];
                const float c11 = cosn[2 - pi][2 - pj];
                // argmin (strict <, first-index ties) then max over the other axis
                const int xi = ((c01 < c00) || (c11 < c10)) ? 1 : 0;
                const int yi = ((c10 < c00) || (c11 < c01)) ? 1 : 0;
                sdx += (float)(xi - pj);
                sdy += (float)(yi - pi);
            }
        }

        const float u = sdx * (1.0f / 436.0f)  + (-1.0f + (float)J * (2.0f / 1023.0f));
        const float v = sdy * (1.0f / 1024.0f) + (-1.0f + (float)I * (2.0f / 435.0f));
        outu[(size_t)I * kW + J] = u;
        outv[(size_t)I * kW + J] = v;
    }
}

extern "C" void kernel_launch(void* const* d_in, const int* in_sizes, int n_in,
                              void* d_out, int out_size, void* d_ws, size_t ws_size,
                              hipStream_t stream) {
    const float* x = (const float*)d_in[0];
    const float* y = (const float*)d_in[1];
    float* out = (float*)d_out;
    dim3 grid(kW / kTJ, (kH + kTI - 1) / kTI, kB);  // (16, 28, 4)
    wdt_flow_kernel<<<grid, kNTH, 0, stream>>>(x, y, out);
}